// FP_Layer_5583457485356
// MI455X (gfx1250) — compile-verified
//
#include <hip/hip_runtime.h>

typedef __attribute__((ext_vector_type(2))) float v2f;
typedef __attribute__((ext_vector_type(8))) float v8f;
typedef __attribute__((ext_vector_type(4))) int   v4i;

#define B_   8
#define N1_  8192
#define N2_  2048
#define C1_  128
#define C2_  256
#define KIN_ 384
#define M_   (B_ * N1_)   // 65536

// gfx1250 async global->LDS path (guarded: falls back to sync staging if the
// toolchain doesn't declare the builtins).
#if defined(__has_builtin)
#  if __has_builtin(__builtin_amdgcn_global_load_async_to_lds_b128) && \
      __has_builtin(__builtin_amdgcn_s_wait_asynccnt)
#    define USE_ASYNC_LDS 1
#  endif
#endif
#ifndef USE_ASYNC_LDS
#  define USE_ASYNC_LDS 0
#endif

#define AS1 __attribute__((address_space(1)))
#define AS3 __attribute__((address_space(3)))

// ---------------------------------------------------------------------------
// 1) KNN: one thread per query point; xyz2 of this batch staged in LDS (24KB)
// ---------------------------------------------------------------------------
__global__ __launch_bounds__(256) void knn_kernel(const float* __restrict__ xyz1,
                                                  const float* __restrict__ xyz2,
                                                  int* __restrict__ idx,
                                                  float* __restrict__ wgt) {
    __shared__ float s[N2_ * 3];
    const int b = blockIdx.y;
    const float* x2 = xyz2 + (size_t)b * N2_ * 3;
    for (int i = threadIdx.x; i < N2_ * 3; i += 256) s[i] = x2[i];
    __syncthreads();

    const int m = blockIdx.x * 256 + threadIdx.x;      // point within batch
    const float* p = xyz1 + ((size_t)b * N1_ + m) * 3;
    const float px = p[0], py = p[1], pz = p[2];

    float d0 = 3.4e38f, d1 = 3.4e38f, d2 = 3.4e38f;
    int   i0 = 0, i1 = 0, i2 = 0;
    for (int j = 0; j < N2_; ++j) {
        const float dx = px - s[3 * j + 0];
        const float dy = py - s[3 * j + 1];
        const float dz = pz - s[3 * j + 2];
        const float dd = dx * dx + dy * dy + dz * dz;
        if (dd < d0)      { d2 = d1; i2 = i1; d1 = d0; i1 = i0; d0 = dd; i0 = j; }
        else if (dd < d1) { d2 = d1; i2 = i1; d1 = dd; i1 = j; }
        else if (dd < d2) { d2 = dd; i2 = j; }
    }
    const float r0 = sqrtf(fmaxf(d0, 0.f));
    const float r1 = sqrtf(fmaxf(d1, 0.f));
    const float r2 = sqrtf(fmaxf(d2, 0.f));
    float w0 = 1.f / fmaxf(r0, 1e-8f);
    float w1 = 1.f / fmaxf(r1, 1e-8f);
    float w2 = 1.f / fmaxf(r2, 1e-8f);
    const float inv = 1.f / (w0 + w1 + w2);
    w0 *= inv; w1 *= inv; w2 *= inv;

    const size_t pm = (size_t)b * N1_ + m;
    idx[pm * 3 + 0] = i0; idx[pm * 3 + 1] = i1; idx[pm * 3 + 2] = i2;
    wgt[pm * 3 + 0] = w0; wgt[pm * 3 + 1] = w1; wgt[pm * 3 + 2] = w2;
}

// ---------------------------------------------------------------------------
// 2) Transpose feats2 (B,C2,N2) -> f2t (B,N2,C2), tiled 32x32 via LDS
// ---------------------------------------------------------------------------
__global__ __launch_bounds__(256) void tr_feats2(const float* __restrict__ f2,
                                                 float* __restrict__ f2t) {
    __shared__ float tile[32][33];
    const int b = blockIdx.z, n0 = blockIdx.x * 32, c0 = blockIdx.y * 32;
    const float* src = f2 + (size_t)b * C2_ * N2_;
    for (int i = threadIdx.y; i < 32; i += 8)
        tile[i][threadIdx.x] = src[(size_t)(c0 + i) * N2_ + n0 + threadIdx.x];
    __syncthreads();
    float* dst = f2t + (size_t)b * N2_ * C2_;
    for (int i = threadIdx.y; i < 32; i += 8)
        dst[(size_t)(n0 + i) * C2_ + c0 + threadIdx.x] = tile[threadIdx.x][i];
}

// ---------------------------------------------------------------------------
// 3) Transpose feats1 (B,C1,N1) into X[:,256:384]   (X is M x 384 row-major)
// ---------------------------------------------------------------------------
__global__ __launch_bounds__(256) void tr_feats1(const float* __restrict__ f1,
                                                 float* __restrict__ X) {
    __shared__ float tile[32][33];
    const int b = blockIdx.z, n0 = blockIdx.x * 32, c0 = blockIdx.y * 32;
    const float* src = f1 + (size_t)b * C1_ * N1_;
    for (int i = threadIdx.y; i < 32; i += 8)
        tile[i][threadIdx.x] = src[(size_t)(c0 + i) * N1_ + n0 + threadIdx.x];
    __syncthreads();
    for (int i = threadIdx.y; i < 32; i += 8)
        X[((size_t)b * N1_ + n0 + i) * KIN_ + 256 + c0 + threadIdx.x] = tile[threadIdx.x][i];
}

// ---------------------------------------------------------------------------
// 4) IDW interpolation into X[:,0:256]; one wave per point, float4 streams
// ---------------------------------------------------------------------------
__global__ __launch_bounds__(256) void interp_kernel(const float* __restrict__ f2t,
                                                     const int* __restrict__ idx,
                                                     const float* __restrict__ wgt,
                                                     float* __restrict__ X) {
    const int wave = threadIdx.x >> 5, lane = threadIdx.x & 31;
    const int m = blockIdx.x * 8 + wave;          // global point id
    const int b = m >> 13;                        // /8192
    const int j0 = idx[(size_t)m * 3 + 0];
    const int j1 = idx[(size_t)m * 3 + 1];
    const int j2 = idx[(size_t)m * 3 + 2];
    const float w0 = wgt[(size_t)m * 3 + 0];
    const float w1 = wgt[(size_t)m * 3 + 1];
    const float w2 = wgt[(size_t)m * 3 + 2];
    const float4* r0 = (const float4*)(f2t + ((size_t)b * N2_ + j0) * C2_);
    const float4* r1 = (const float4*)(f2t + ((size_t)b * N2_ + j1) * C2_);
    const float4* r2 = (const float4*)(f2t + ((size_t)b * N2_ + j2) * C2_);
    float4* xr = (float4*)(X + (size_t)m * KIN_);
    for (int c = lane; c < C2_ / 4; c += 32) {
        const float4 a = r0[c], bb = r1[c], cc = r2[c];
        float4 o;
        o.x = w0 * a.x + w1 * bb.x + w2 * cc.x;
        o.y = w0 * a.y + w1 * bb.y + w2 * cc.y;
        o.z = w0 * a.z + w1 * bb.z + w2 * cc.z;
        o.w = w0 * a.w + w1 * bb.w + w2 * cc.w;
        xr[c] = o;
    }
}

// ---------------------------------------------------------------------------
// 5) fp32 WMMA GEMM:  D(MxN) = A(MxK) * W(NxK)^T + bias,   optional BN+ReLU
//    applied to A elements on load (channel = K index).
//    Block: 256 threads = 8 waves arranged WM x WN; wave tile = 16M x 64N.
//    Tiles staged to LDS with GLOBAL_LOAD_ASYNC_TO_LDS_B128 when available.
//    LDS rows padded to 36 floats: 16B-aligned rows, half-wave K-offsets (+2)
//    land in disjoint bank sets (36*l mod 64 even, +2 odd-shifted).
// ---------------------------------------------------------------------------
template <int KDIM, int NDIM, int WM, int WN, bool BNRELU>
__global__ __launch_bounds__(256) void gemm_wmma(const float* __restrict__ A,
                                                 const float* __restrict__ W,
                                                 const float* __restrict__ bias,
                                                 const float* __restrict__ scale,
                                                 const float* __restrict__ shift,
                                                 float* __restrict__ D) {
    constexpr int KS = 32;
    constexpr int MT = 16 * WM;
    constexpr int LP = KS + 4;            // padded row length (36 floats, 144B)
    __shared__ float Xs[MT][LP];          // A tile, [row][k]
    __shared__ float Ws[NDIM][LP];        // B tile as W rows, [n][k]

    const int t = threadIdx.x;
    const int wave = t >> 5, lane = t & 31;
    const int wm = wave % WM, wn = wave / WM;
    const int m0 = blockIdx.x * MT;
    const int mrow = wm * 16 + (lane & 15);
    const int khalf = (lane >> 4) * 2;    // 0 or 2: K-pair owned by this half-wave
    const int col16 = lane & 15;

    v8f acc[4] = {};

    for (int k0 = 0; k0 < KDIM; k0 += KS) {
        // ---- stage W tile: Ws[n][k] = W[n][k0+k] (16B chunks, contiguous) ----
#if USE_ASYNC_LDS
        for (int q = t; q < NDIM * (KS / 4); q += 256) {
            const int n = q >> 3, cw = (q & 7) * 4;
            __builtin_amdgcn_global_load_async_to_lds_b128(
                (AS1 v4i*)(W + (size_t)n * KDIM + k0 + cw),
                (AS3 v4i*)&Ws[n][cw], 0, 0);
        }
#else
        for (int q = t; q < NDIM * (KS / 4); q += 256) {
            const int n = q >> 3, cw = (q & 7) * 4;
            const float4 v = *(const float4*)(W + (size_t)n * KDIM + k0 + cw);
            *(float4*)&Ws[n][cw] = v;
        }
#endif
        // ---- stage A tile ----
        if (BNRELU) {
            // fold BN1 + ReLU into the load (per-K-channel affine)
            for (int i = t; i < MT * KS; i += 256) {
                const int r = i >> 5, c = i & (KS - 1);
                float v = A[(size_t)(m0 + r) * KDIM + k0 + c];
                const int ch = k0 + c;
                Xs[r][c] = fmaxf(v * scale[ch] + shift[ch], 0.f);
            }
        } else {
#if USE_ASYNC_LDS
            for (int q = t; q < MT * (KS / 4); q += 256) {
                const int r = q >> 3, cw = (q & 7) * 4;
                __builtin_amdgcn_global_load_async_to_lds_b128(
                    (AS1 v4i*)(A + (size_t)(m0 + r) * KDIM + k0 + cw),
                    (AS3 v4i*)&Xs[r][cw], 0, 0);
            }
#else
            for (int q = t; q < MT * (KS / 4); q += 256) {
                const int r = q >> 3, cw = (q & 7) * 4;
                const float4 v = *(const float4*)(A + (size_t)(m0 + r) * KDIM + k0 + cw);
                *(float4*)&Xs[r][cw] = v;
            }
#endif
        }
#if USE_ASYNC_LDS
        __builtin_amdgcn_s_wait_asynccnt(0);
#endif
        __syncthreads();

        // ---- compute: 8 K-substeps x 4 N-subtiles of fp32 WMMA ----
#pragma unroll
        for (int kk = 0; kk < KS; kk += 4) {
            const v2f a = *(const v2f*)&Xs[mrow][kk + khalf];
#pragma unroll
            for (int j = 0; j < 4; ++j) {
                const int n = wn * 64 + j * 16 + col16;
                const v2f bb = *(const v2f*)&Ws[n][kk + khalf];
                acc[j] = __builtin_amdgcn_wmma_f32_16x16x4_f32(
                    false, a, false, bb, (short)0, acc[j], false, false);
            }
        }
        __syncthreads();
    }

    // epilogue: D row = m0 + wm*16 + i + 8*(lane>>4), col = wn*64 + j*16 + (lane&15)
    const int rowbase = m0 + wm * 16 + (lane >> 4) * 8;
#pragma unroll
    for (int j = 0; j < 4; ++j) {
        const int col = wn * 64 + j * 16 + col16;
        const float bc = bias[col];
#pragma unroll
        for (int i = 0; i < 8; ++i)
            D[(size_t)(rowbase + i) * NDIM + col] = acc[j][i] + bc;
    }
}

// ---------------------------------------------------------------------------
// 6) Column stats (deterministic two-pass): per-block partial sum/sumsq
// ---------------------------------------------------------------------------
template <int C>
__global__ void colstats_partial(const float* __restrict__ H, float* __restrict__ part) {
    const int c = threadIdx.x;
    const int rows = M_ / 256;
    const int r0 = blockIdx.x * rows;
    float s = 0.f, s2 = 0.f;
    for (int r = r0; r < r0 + rows; ++r) {
        const float v = H[(size_t)r * C + c];
        s += v; s2 += v * v;
    }
    part[(size_t)blockIdx.x * 2 * C + c] = s;
    part[(size_t)blockIdx.x * 2 * C + C + c] = s2;
}

template <int C>
__global__ void bn_finalize(const float* __restrict__ part,
                            const float* __restrict__ g, const float* __restrict__ be,
                            float* __restrict__ scale, float* __restrict__ shift) {
    const int c = threadIdx.x;
    float s = 0.f, s2 = 0.f;
    for (int b = 0; b < 256; ++b) {           // fixed order -> deterministic
        s  += part[(size_t)b * 2 * C + c];
        s2 += part[(size_t)b * 2 * C + C + c];
    }
    const float mean = s * (1.f / M_);
    const float var  = fmaxf(s2 * (1.f / M_) - mean * mean, 0.f);
    const float sc = g[c] * rsqrtf(var + 1e-5f);
    scale[c] = sc;
    shift[c] = be[c] - mean * sc;
}

// ---------------------------------------------------------------------------
// 7) BN2+ReLU + transpose H2 (M x 128) -> out (B, 128, N1)
// ---------------------------------------------------------------------------
__global__ __launch_bounds__(256) void out_transpose(const float* __restrict__ H2,
                                                     const float* __restrict__ scale,
                                                     const float* __restrict__ shift,
                                                     float* __restrict__ out) {
    __shared__ float tile[32][33];
    const int b = blockIdx.z, n0 = blockIdx.x * 32, c0 = blockIdx.y * 32;
    for (int i = threadIdx.y; i < 32; i += 8) {
        const int n = n0 + i, c = c0 + threadIdx.x;
        const float v = H2[((size_t)b * N1_ + n) * C1_ + c];
        tile[i][threadIdx.x] = fmaxf(v * scale[c] + shift[c], 0.f);
    }
    __syncthreads();
    for (int i = threadIdx.y; i < 32; i += 8) {
        const int c = c0 + i, n = n0 + threadIdx.x;
        out[((size_t)b * C1_ + c) * N1_ + n] = tile[threadIdx.x][i];
    }
}

// ---------------------------------------------------------------------------
static inline size_t alignUp(size_t x) { return (x + 255) & ~(size_t)255; }

extern "C" void kernel_launch(void* const* d_in, const int* in_sizes, int n_in,
                              void* d_out, int out_size, void* d_ws, size_t ws_size,
                              hipStream_t stream) {
    (void)in_sizes; (void)n_in; (void)out_size; (void)ws_size;

    const float* xyz1   = (const float*)d_in[0];
    const float* xyz2   = (const float*)d_in[1];
    const float* feats1 = (const float*)d_in[2];
    const float* feats2 = (const float*)d_in[3];
    const float* W1     = (const float*)d_in[4];
    const float* b1     = (const float*)d_in[5];
    const float* g1     = (const float*)d_in[6];
    const float* be1    = (const float*)d_in[7];
    const float* W2     = (const float*)d_in[8];
    const float* b2     = (const float*)d_in[9];
    const float* g2     = (const float*)d_in[10];
    const float* be2    = (const float*)d_in[11];
    float* out = (float*)d_out;

    char* ws = (char*)d_ws;
    size_t off = 0;
    auto take = [&](size_t bytes) { void* p = ws + off; off = alignUp(off + bytes); return p; };

    int*   idx    = (int*)  take((size_t)M_ * 3 * 4);
    float* wgt    = (float*)take((size_t)M_ * 3 * 4);
    float* f2t    = (float*)take((size_t)B_ * N2_ * C2_ * 4);
    float* X      = (float*)take((size_t)M_ * KIN_ * 4);
    float* H1     = (float*)take((size_t)M_ * 256 * 4);
    float* H2     = (float*)take((size_t)M_ * 128 * 4);
    float* part1  = (float*)take((size_t)256 * 2 * 256 * 4);
    float* part2  = (float*)take((size_t)256 * 2 * 128 * 4);
    float* scale1 = (float*)take(256 * 4);
    float* shift1 = (float*)take(256 * 4);
    float* scale2 = (float*)take(128 * 4);
    float* shift2 = (float*)take(128 * 4);

    // 1) KNN + weights
    knn_kernel<<<dim3(N1_ / 256, B_), 256, 0, stream>>>(xyz1, xyz2, idx, wgt);
    // 2) feats2 transpose
    tr_feats2<<<dim3(N2_ / 32, C2_ / 32, B_), dim3(32, 8), 0, stream>>>(feats2, f2t);
    // 3) feats1 -> X[:,256:384]
    tr_feats1<<<dim3(N1_ / 32, C1_ / 32, B_), dim3(32, 8), 0, stream>>>(feats1, X);
    // 4) interpolation -> X[:,0:256]
    interp_kernel<<<dim3(M_ / 8), 256, 0, stream>>>(f2t, idx, wgt, X);
    // 5) GEMM1: H1 = X * W1^T + b1     (M x 384) * (384 x 256)
    gemm_wmma<KIN_, 256, 2, 4, false><<<dim3(M_ / 32), 256, 0, stream>>>(
        X, W1, b1, nullptr, nullptr, H1);
    // 6) BN1 stats
    colstats_partial<256><<<dim3(256), 256, 0, stream>>>(H1, part1);
    bn_finalize<256><<<dim3(1), 256, 0, stream>>>(part1, g1, be1, scale1, shift1);
    // 7) GEMM2: H2 = relu(BN1(H1)) * W2^T + b2    (M x 256) * (256 x 128)
    gemm_wmma<256, 128, 4, 2, true><<<dim3(M_ / 64), 256, 0, stream>>>(
        H1, W2, b2, scale1, shift1, H2);
    // 8) BN2 stats
    colstats_partial<128><<<dim3(256), 128, 0, stream>>>(H2, part2);
    bn_finalize<128><<<dim3(1), 128, 0, stream>>>(part2, g2, be2, scale2, shift2);
    // 9) BN2+ReLU + transpose to (B, 128, N1)
    out_transpose<<<dim3(N1_ / 32, 128 / 32, B_), dim3(32, 8), 0, stream>>>(
        H2, scale2, shift2, out);
}